// GraphAttentionLayer_38654705664189
// MI455X (gfx1250) — compile-verified
//
#include <hip/hip_runtime.h>
#include <hip/hip_bf16.h>

#define ALPHA 0.2f

typedef __attribute__((ext_vector_type(2))) float v2f;
typedef __attribute__((ext_vector_type(8))) float v8f;

static constexpr int B = 8;
static constexpr int N = 2048;
static constexpr int F = 256;        // F_in == F_out
static constexpr int ROWS = B * N;   // 16384
static constexpr int NT = N / 16;    // 128 j-tiles

// gfx1250 async global->LDS copy (ASYNCcnt-tracked, CDNA5 ISA).
// %0 = per-lane LDS byte offset (VDST), %1 = 64-bit global address (VADDR).
__device__ __forceinline__ void async_copy_b128(unsigned lds_off, const float* gptr) {
  asm volatile("global_load_async_to_lds_b128 %0, %1, off"
               :: "v"(lds_off), "v"((unsigned long long)(uintptr_t)gptr)
               : "memory");
}

// ---------------- Kernel 1: Wh = h @ W  (fp32 WMMA 16x16x4) ----------------
__global__ void __launch_bounds__(32)
gat_gemm_wh(const float* __restrict__ h, const float* __restrict__ W,
            float* __restrict__ Wh) {
  const int lane = threadIdx.x & 31;
  const int m    = lane & 15;
  const int half = lane >> 4;
  const int c0   = blockIdx.x * 16;
  const int r0   = blockIdx.y * 16;

  v8f acc = {};
  const float* arow = h + (size_t)(r0 + m) * F;
  for (int k0 = 0; k0 < F; k0 += 4) {
    const float2 av = *reinterpret_cast<const float2*>(arow + k0 + 2 * half);
    v2f a; a.x = av.x; a.y = av.y;
    const int kr = k0 + 2 * half;
    v2f bf;
    bf.x = W[(size_t)kr * F + c0 + m];
    bf.y = W[(size_t)(kr + 1) * F + c0 + m];
    acc = __builtin_amdgcn_wmma_f32_16x16x4_f32(false, a, false, bf,
                                                (short)0, acc, false, false);
  }
#pragma unroll
  for (int r = 0; r < 8; ++r)
    Wh[(size_t)(r0 + r + 8 * half) * F + c0 + m] = acc[r];
}

// ---------------- Kernel 2: e_src/e_dst = Wh @ a_src / a_dst ----------------
__global__ void __launch_bounds__(256)
gat_edge(const float* __restrict__ Wh, const float* __restrict__ a,
         float* __restrict__ e_src, float* __restrict__ e_dst) {
  const int lane = threadIdx.x & 31;
  const int wave = threadIdx.x >> 5;
  const int row  = blockIdx.x * 8 + wave;
  const float* wr = Wh + (size_t)row * F;
  float ps = 0.f, pd = 0.f;
#pragma unroll
  for (int i = 0; i < 8; ++i) {
    const int c = i * 32 + lane;
    const float v = wr[c];
    ps += v * a[c];
    pd += v * a[F + c];
  }
#pragma unroll
  for (int off = 16; off >= 1; off >>= 1) {
    ps += __shfl_xor(ps, off, 32);
    pd += __shfl_xor(pd, off, 32);
  }
  if (lane == 0) { e_src[row] = ps; e_dst[row] = pd; }
}

// ---------------- Kernel 3: per-batch max of e_dst ----------------
__global__ void __launch_bounds__(256)
gat_dmax(const float* __restrict__ e_dst, float* __restrict__ dmax) {
  __shared__ float red[256];
  const int b = blockIdx.x;
  float mv = -3.402823466e38f;
  for (int j = threadIdx.x; j < N; j += 256)
    mv = fmaxf(mv, e_dst[(size_t)b * N + j]);
  red[threadIdx.x] = mv;
  __syncthreads();
  for (int s = 128; s > 0; s >>= 1) {
    if (threadIdx.x < s)
      red[threadIdx.x] = fmaxf(red[threadIdx.x], red[threadIdx.x + s]);
    __syncthreads();
  }
  if (threadIdx.x == 0) dmax[b] = red[0];
}

// ---------------- Kernel 4: per-row softmax stats ----------------
__global__ void __launch_bounds__(256)
gat_rowstats(const float* __restrict__ e_src, const float* __restrict__ e_dst,
             const float* __restrict__ dmax, float* __restrict__ m_row,
             float* __restrict__ inv_l) {
  const int lane = threadIdx.x & 31;
  const int wave = threadIdx.x >> 5;
  const int row  = blockIdx.x * 8 + wave;
  const int b    = row / N;
  const float es = e_src[row];
  float mrow = es + dmax[b];
  mrow = mrow > 0.f ? mrow : ALPHA * mrow;
  float l = 0.f;
  const float* ed = e_dst + (size_t)b * N;
  for (int j = lane; j < N; j += 32) {
    float x = es + ed[j];
    x = x > 0.f ? x : ALPHA * x;
    l += __expf(x - mrow);
  }
#pragma unroll
  for (int off = 16; off >= 1; off >>= 1) l += __shfl_xor(l, off, 32);
  if (lane == 0) { m_row[row] = mrow; inv_l[row] = 1.0f / l; }
}

// ---------------- Kernel 5: h' = softmax(e) @ Wh  (fp32 WMMA) ----------------
// Block = 16 waves = one (batch, 16-row i-tile). Wh j-tiles (16x256) are
// double-buffered in LDS via async global->LDS b128 copies issued one tile
// ahead; P tile (16x16 probs) built once per j-tile and shared by all waves.
__global__ void __launch_bounds__(512)
gat_attn(const float* __restrict__ Wh, const float* __restrict__ e_src,
         const float* __restrict__ e_dst, const float* __restrict__ m_row,
         const float* __restrict__ inv_l, float* __restrict__ out) {
  __shared__ float WhT[2][16 * F];        // 2 x 16KB double buffer
  __shared__ float Pl[16 * 16];           // Pl[k*16 + m] = P[m, k]
  __shared__ float es_t[16], mr_t[16], il_t[16];

  const int tid  = threadIdx.x;
  const int lane = tid & 31;
  const int wave = tid >> 5;              // 0..15 -> output col tile
  const int m    = lane & 15;
  const int half = lane >> 4;
  const int b    = blockIdx.y;
  const int i0   = blockIdx.x * 16;
  const int rowbase = b * N + i0;
  const int c0   = wave * 16;

  // LDS byte offset of the staging buffer (generic LDS addr low 32 bits).
  const unsigned lds_wht =
      (unsigned)(uintptr_t)(__attribute__((address_space(3))) float*)(&WhT[0][0]);

  if (tid < 16) {
    es_t[tid] = e_src[rowbase + tid];
    mr_t[tid] = m_row[rowbase + tid];
    il_t[tid] = inv_l[rowbase + tid];
  }

  const float* whb = Wh + (size_t)b * N * F;
  const float* edb = e_dst + (size_t)b * N;

  // Prologue: stage j-tile 0 (16x256 floats; 512 thr x 2 x b128).
  async_copy_b128(lds_wht + tid * 16,        whb + tid * 4);
  async_copy_b128(lds_wht + 8192 + tid * 16, whb + 2048 + tid * 4);
  __syncthreads();   // es_t/mr_t/il_t visible

  v8f acc = {};
  for (int jt = 0; jt < NT; ++jt) {
    const int j0  = jt * 16;
    const int buf = jt & 1;

    if (tid < 256) {                      // build 16x16 P tile
      const int k = tid >> 4, mm = tid & 15;
      float x = es_t[mm] + edb[j0 + k];
      x = x > 0.f ? x : ALPHA * x;
      Pl[k * 16 + mm] = __expf(x - mr_t[mm]) * il_t[mm];
    }

    if (jt + 1 < NT) {                    // prefetch next Wh j-tile
      const float* g = whb + (size_t)(j0 + 16) * F;
      const unsigned dst = lds_wht + ((unsigned)((jt + 1) & 1)) * 16384u + tid * 16;
      async_copy_b128(dst,        g + tid * 4);
      async_copy_b128(dst + 8192, g + 2048 + tid * 4);
      // async loads retire in order: <=2 outstanding means current tile landed
      asm volatile("s_wait_asynccnt 0x2" ::: "memory");
    } else {
      asm volatile("s_wait_asynccnt 0x0" ::: "memory");
    }
    __syncthreads();                      // P tile + Wh tile ready for all waves

    const float* wtile = &WhT[buf][0];
#pragma unroll
    for (int kk = 0; kk < 4; ++kk) {
      v2f a;
      a.x = Pl[(kk * 4 + 2 * half) * 16 + m];
      a.y = Pl[(kk * 4 + 2 * half + 1) * 16 + m];
      const int kr = kk * 4 + 2 * half;
      v2f bf;
      bf.x = wtile[kr * F + c0 + m];
      bf.y = wtile[(kr + 1) * F + c0 + m];
      acc = __builtin_amdgcn_wmma_f32_16x16x4_f32(false, a, false, bf,
                                                  (short)0, acc, false, false);
    }
    __syncthreads();                      // protect Pl before next overwrite
  }
#pragma unroll
  for (int r = 0; r < 8; ++r)
    out[(size_t)(b * N + i0 + r + 8 * half) * F + c0 + m] = acc[r];
}

extern "C" void kernel_launch(void* const* d_in, const int* in_sizes, int n_in,
                              void* d_out, int out_size, void* d_ws, size_t ws_size,
                              hipStream_t stream) {
  const float* h = (const float*)d_in[0];   // (8, 2048, 256) f32
  const float* W = (const float*)d_in[1];   // (256, 256) f32
  const float* a = (const float*)d_in[2];   // (512,) f32
  float* out = (float*)d_out;               // (8, 2048, 256) f32

  char* ws = (char*)d_ws;
  float* Wh    = (float*)ws;                         // 16 MiB
  float* e_src = (float*)(ws + (size_t)ROWS * F * sizeof(float));
  float* e_dst = e_src + ROWS;
  float* dmax  = e_dst + ROWS;
  float* m_row = dmax + 16;
  float* inv_l = m_row + ROWS;

  gat_gemm_wh <<<dim3(F / 16, ROWS / 16), 32, 0, stream>>>(h, W, Wh);
  gat_edge    <<<ROWS / 8, 256, 0, stream>>>(Wh, a, e_src, e_dst);
  gat_dmax    <<<B, 256, 0, stream>>>(e_dst, dmax);
  gat_rowstats<<<ROWS / 8, 256, 0, stream>>>(e_src, e_dst, dmax, m_row, inv_l);
  gat_attn    <<<dim3(N / 16, B), 512, 0, stream>>>(Wh, e_src, e_dst, m_row, inv_l, out);
}